// Sparsegen_lin_61856118997451
// MI455X (gfx1250) — compile-verified
//
#include <hip/hip_runtime.h>

// sparsegen-lin over rows of [16384, 4096] fp32.
// tau solves sum(relu(z - tau)) = 1 - LAM  (exact equivalent of the sort path).
// One workgroup per row; TDM DMAs the row into LDS; Newton root-find; single
// HBM pass (read row once, write row once, nontemporal stores).

#define LAM          0.1f
#define DIM          4096
#define THREADS      256
#define NEWTON_ITERS 10

typedef float        f32x4 __attribute__((ext_vector_type(4)));
typedef unsigned int u32x4 __attribute__((ext_vector_type(4)));
typedef int          i32x8 __attribute__((ext_vector_type(8)));
typedef int          i32x4 __attribute__((ext_vector_type(4)));

#define ACC1(x)            \
  do {                     \
    if ((x) > tau) {       \
      s += (x);            \
      c += 1.0f;           \
    }                      \
  } while (0)
#define ACC4(v) do { ACC1((v).x); ACC1((v).y); ACC1((v).z); ACC1((v).w); } while (0)

__global__ __launch_bounds__(THREADS) void sparsegen_rows(const float* __restrict__ in,
                                                          float* __restrict__ out) {
  __shared__ float smem[DIM];   // one full row, 16 KB
  __shared__ float red_s[8];
  __shared__ float red_c[8];
  __shared__ float red_m[8];

  const int tid  = threadIdx.x;
  const int lane = tid & 31;
  const int wid  = tid >> 5;
  const unsigned long long row = blockIdx.x;

  const float* __restrict__ src = in  + row * (unsigned long long)DIM;
  float*       __restrict__ dst = out + row * (unsigned long long)DIM;

  // ---- Tensor Data Mover: DMA this row (4096 x fp32 = 16 KB) into LDS ----
  // D# per CDNA5 ISA ch.8: group0 = {flags, lds_addr, global_addr, type},
  // group1 = dims/strides for a 2D tile (tile_dim0=4096, tile_dim1=1).
  if (wid == 0) {
    unsigned lds_addr      = (unsigned)(size_t)(void*)smem;       // LDS aperture low 32b = LDS byte offset
    unsigned long long ga  = (unsigned long long)(const void*)src;

    u32x4 g0;
    g0[0] = 1u;                                                   // count=1 (valid), user mode, no gather
    g0[1] = lds_addr;                                             // lds_addr[31:0]
    g0[2] = (unsigned)ga;                                         // global_addr[31:0]
    g0[3] = (unsigned)((ga >> 32) & 0x01FFFFFFull) | 0x80000000u; // global_addr[56:32] | type=2

    i32x8 g1;
    g1[0] = (int)(2u << 16);                            // workgroup_mask=0, data_size=2 (4 bytes)
    g1[1] = (int)(((unsigned)DIM & 0xFFFFu) << 16);     // tensor_dim0[15:0] in [31:16]
    g1[2] = (int)(1u << 16);                            // tensor_dim0[31:16]=0, tensor_dim1[15:0]=1
    g1[3] = (int)(((unsigned)DIM & 0xFFFFu) << 16);     // tensor_dim1[31:16]=0, tile_dim0=4096
    g1[4] = 1;                                          // tile_dim1=1, tile_dim2=0
    g1[5] = DIM;                                        // tensor_dim0_stride[31:0]=4096
    g1[6] = 0;                                          // stride hi / dim1_stride lo
    g1[7] = 0;

    i32x4 gz4 = {0, 0, 0, 0};                           // groups 2/3 unused (<=2D)
    i32x8 gz8 = {0, 0, 0, 0, 0, 0, 0, 0};
    __builtin_amdgcn_tensor_load_to_lds(g0, g1, gz4, gz4, gz8, 0);
    __builtin_amdgcn_s_wait_tensorcnt(0);
  }
  __syncthreads();

  // ---- 16 values per thread, interleaved float4 layout (bank-conflict-free) ----
  const f32x4* sm4 = (const f32x4*)smem;
  f32x4 v0 = sm4[tid];
  f32x4 v1 = sm4[tid + 256];
  f32x4 v2 = sm4[tid + 512];
  f32x4 v3 = sm4[tid + 768];

  // ---- row max (wave32 butterfly + 8-wave LDS combine) ----
  float tmax = fmaxf(fmaxf(fmaxf(v0.x, v0.y), fmaxf(v0.z, v0.w)),
                     fmaxf(fmaxf(fmaxf(v1.x, v1.y), fmaxf(v1.z, v1.w)),
                           fmaxf(fmaxf(fmaxf(v2.x, v2.y), fmaxf(v2.z, v2.w)),
                                 fmaxf(fmaxf(v3.x, v3.y), fmaxf(v3.z, v3.w)))));
  float m = tmax;
#pragma unroll
  for (int o = 16; o > 0; o >>= 1) m = fmaxf(m, __shfl_xor(m, o, 32));
  if (lane == 0) red_m[wid] = m;
  __syncthreads();
  float rowmax = red_m[0];
#pragma unroll
  for (int w = 1; w < 8; ++w) rowmax = fmaxf(rowmax, red_m[w]);

  // ---- Newton root-find for tau: f(tau) = sum(relu(z - tau)) = 1 - LAM ----
  // tau increases monotonically => support shrinks monotonically => threads whose
  // local max fell below tau skip the scan (whole waves drop out via execz).
  const float target = 1.0f - LAM;
  float tau = rowmax - target;   // f(tau0) >= target, so tau0 <= tau*

  for (int it = 0; it < NEWTON_ITERS; ++it) {
    float s = 0.0f, c = 0.0f;
    if (tmax > tau) {
      ACC4(v0); ACC4(v1); ACC4(v2); ACC4(v3);
    }
#pragma unroll
    for (int o = 16; o > 0; o >>= 1) {
      s += __shfl_xor(s, o, 32);
      c += __shfl_xor(c, o, 32);
    }
    if (lane == 0) { red_s[wid] = s; red_c[wid] = c; }
    __syncthreads();
    float S = 0.0f, C = 0.0f;
#pragma unroll
    for (int w = 0; w < 8; ++w) { S += red_s[w]; C += red_c[w]; }
    __syncthreads();                 // protect red_s/red_c before next iteration
    tau = (S - target) / C;          // C >= 1 always (tau < rowmax invariant)
  }

  // ---- emit prob = relu(z - tau) / (1 - LAM), streaming stores ----
  const float inv = 1.0f / (1.0f - LAM);
  f32x4* d4 = (f32x4*)dst;
#define EMIT(v, idx)                                              \
  do {                                                            \
    f32x4 r;                                                      \
    r.x = fmaxf((v).x - tau, 0.0f) * inv;                         \
    r.y = fmaxf((v).y - tau, 0.0f) * inv;                         \
    r.z = fmaxf((v).z - tau, 0.0f) * inv;                         \
    r.w = fmaxf((v).w - tau, 0.0f) * inv;                         \
    __builtin_nontemporal_store(r, &d4[idx]);                     \
  } while (0)
  EMIT(v0, tid);
  EMIT(v1, tid + 256);
  EMIT(v2, tid + 512);
  EMIT(v3, tid + 768);
#undef EMIT
}

extern "C" void kernel_launch(void* const* d_in, const int* in_sizes, int n_in,
                              void* d_out, int out_size, void* d_ws, size_t ws_size,
                              hipStream_t stream) {
  const float* in = (const float*)d_in[0];
  float* out = (float*)d_out;
  const int rows = in_sizes[0] / DIM;   // 16384
  sparsegen_rows<<<dim3(rows), dim3(THREADS), 0, stream>>>(in, out);
}